// SelfAttention_4647154614504
// MI455X (gfx1250) — compile-verified
//
#include <hip/hip_runtime.h>
#include <math.h>

// ---------------------------------------------------------------------------
// Self-attention (bidirectional encoder attention), MI455X / gfx1250.
//   query  = input(32x1024) @ Wq^T(1024x1024) + bq          -> WMMA f32 GEMM
//   scores = softmax_t( <query_b, keys_{b,t}> / sqrt(512) ) -> flash partials
//   values = sum_t scores_{b,t} * keys_{b,t}                -> fused
// keys(b,t,dir*512+h) = enc_h[dir][t][b][h]  (contiguous in h -> coalesced)
//
// Roofline: enc_h = 256 MB f32 dominates; ~0.34 GFLOP total => ~1.3 FLOP/byte
// => bandwidth bound; floor ~11 us at 23.3 TB/s with a single enc_h pass.
// Phase B re-reads each 1 MB chunk via async-DMA LDS tiles (L2-resident).
// ---------------------------------------------------------------------------

typedef __attribute__((ext_vector_type(2))) float v2f;
typedef __attribute__((ext_vector_type(8))) float v8f;
typedef __attribute__((ext_vector_type(4))) int   v4i;

#define B_   32
#define T_   2048
#define C_   1024      // input feature size (== D*H here)
#define DH_  1024      // D*H
#define H_   512
#define NCH  8         // t-chunks per batch
#define TC   (T_ / NCH) // 256 timesteps per chunk
#define TT   8          // timesteps per LDS tile in phase B
#define SCALE 0.04419417382415922f  // 1/sqrt(512)

// Workspace layout (floats)
#define Q_OFF   0                      // 32*1024      = 32768
#define S_OFF   32768                  // 32*2048      = 65536 (raw scaled scores)
#define M_OFF   98304                  // 32*8 chunk maxima
#define SU_OFF  98560                  // 32*8 chunk exp-sums
#define P_OFF   98816                  // 32*8*1024 partial weighted values

// ---- CDNA5 async global->LDS DMA (guarded: fall back to direct loads) -----
#if __has_builtin(__builtin_amdgcn_global_load_async_to_lds_b128)
#define HAVE_ASYNC_LDS 1
#else
#define HAVE_ASYNC_LDS 0
#endif

#if HAVE_ASYNC_LDS
// Param 1 of the builtin is 'int4 addrspace(1)*' (from clang diagnostic);
// param 2 is the LDS destination, addrspace(3).
typedef __attribute__((address_space(1))) v4i gv4i;
typedef __attribute__((address_space(3))) v4i lv4i;

static __device__ __forceinline__ void async_wait0() {
#if __has_builtin(__builtin_amdgcn_s_wait_asynccnt)
  __builtin_amdgcn_s_wait_asynccnt(0);
#else
  asm volatile("s_wait_asynccnt 0" ::: "memory");
#endif
}
#endif

// ---------------------------------------------------------------------------
// Kernel 1: Q = input @ Wq^T + bq via V_WMMA_F32_16X16X4_F32.
// 2 row-tiles x 64 col-tiles = 128 tiles, one wave32 per 16x16 tile.
// A fragment (16x4): lane holds row M=lane&15; VGPR pair covers K = 2*(lane>>4)+{0,1}.
// B fragment (4x16): B[k][n] = Wq[n][k]; lane holds col N=lane&15, same K split.
// D/C (16x16 f32): VGPR v -> row M = 8*(lane>>4)+v, col N = lane&15.
// ---------------------------------------------------------------------------
__global__ void qproj_wmma(const float* __restrict__ inp,
                           const float* __restrict__ Wq,
                           const float* __restrict__ bq,
                           float* __restrict__ Q) {
  const int wave = threadIdx.x >> 5;
  const int lane = threadIdx.x & 31;
  const int tile = blockIdx.x * (blockDim.x >> 5) + wave;
  const int mt   = tile & 1;    // row tile (B=32 -> 2 tiles)
  const int nt   = tile >> 1;   // col tile (1024 -> 64 tiles)
  const int half = lane >> 4;
  const int l16  = lane & 15;
  const int arow = mt * 16 + l16;
  const int bcol = nt * 16 + l16;

  const float* __restrict__ ap = inp + (size_t)arow * C_;
  const float* __restrict__ bp = Wq  + (size_t)bcol * C_;

  v8f c = {};
  for (int k0 = 0; k0 < C_; k0 += 4) {
    const int kb = k0 + half * 2;
    v2f a, b;
    a.x = ap[kb]; a.y = ap[kb + 1];
    b.x = bp[kb]; b.y = bp[kb + 1];
    c = __builtin_amdgcn_wmma_f32_16x16x4_f32(
        /*neg_a=*/false, a, /*neg_b=*/false, b,
        /*c_mod=*/(short)0, c, /*reuse_a=*/false, /*reuse_b=*/false);
  }

  const float bias = bq[bcol];
#pragma unroll
  for (int v = 0; v < 8; ++v) {
    const int row = mt * 16 + half * 8 + v;
    Q[(size_t)row * DH_ + bcol] = c[v] + bias;
  }
}

// ---------------------------------------------------------------------------
// Kernel 2: per (chunk, batch) flash partial.
//   Phase A: raw scores for 256 t's (16 waves, one dot per wave iter,
//            coalesced over h, butterfly shfl reduction).
//   Phase B: chunk-local softmax weights + partial weighted value vector.
//            Keys are staged through double-buffered LDS tiles via
//            GLOBAL_LOAD_ASYNC_TO_LDS_B128 (ASYNCcnt), overlapping DMA of
//            tile i+1 with VALU accumulation from tile i.
// ---------------------------------------------------------------------------
__global__ void attn_partial(const float* __restrict__ enc,
                             const float* __restrict__ Q,
                             float* __restrict__ Sraw,
                             float* __restrict__ Mc,
                             float* __restrict__ Sc,
                             float* __restrict__ P) {
  const int ch   = blockIdx.x;
  const int b    = blockIdx.y;
  const int tid  = threadIdx.x;    // 0..511
  const int wave = tid >> 5;       // 0..15
  const int lane = tid & 31;

  __shared__ float qs[DH_];    // query vector for this batch (4 KB)
  __shared__ float sc[TC];     // raw (scaled) scores for this chunk (1 KB)
  __shared__ float wts[TC];    // exp(sc - m_local) (1 KB)
  __shared__ float red[16];
#if HAVE_ASYNC_LDS
  __shared__ float ktile[2][TT * 2 * H_];  // 2 x 32 KB key tiles
#endif

  qs[tid]       = Q[(size_t)b * DH_ + tid];
  qs[tid + 512] = Q[(size_t)b * DH_ + tid + 512];
  __syncthreads();

  // ---- Phase A: scores --------------------------------------------------
  float wmax = -INFINITY;
  for (int i = 0; i < TC / 16; ++i) {
    const int tt = wave + 16 * i;       // local t
    const int t  = ch * TC + tt;        // global t
    const float* __restrict__ k0p = enc + ((size_t)t * B_ + b) * H_;
    const float* __restrict__ k1p = enc + (((size_t)T_ + t) * B_ + b) * H_;
    float acc = 0.f;
#pragma unroll
    for (int j = 0; j < H_ / 32; ++j) {
      const int h = lane + 32 * j;
      acc += qs[h] * k0p[h];
      acc += qs[H_ + h] * k1p[h];
    }
#pragma unroll
    for (int off = 16; off > 0; off >>= 1)
      acc += __shfl_xor(acc, off, 32);
    const float s = acc * SCALE;        // uniform across wave after butterfly
    if (lane == 0) sc[tt] = s;
    wmax = fmaxf(wmax, s);
  }
  if (lane == 0) red[wave] = wmax;
  __syncthreads();
  if (tid == 0) {
    float m = red[0];
    for (int w = 1; w < 16; ++w) m = fmaxf(m, red[w]);
    red[0] = m;
  }
  __syncthreads();
  const float m_local = red[0];
  __syncthreads();

  // ---- chunk-local weights + sum, emit raw scores -----------------------
  float psum = 0.f;
  if (tid < TC) {
    const float w = __expf(sc[tid] - m_local);
    wts[tid] = w;
    psum = w;
    Sraw[(size_t)b * T_ + ch * TC + tid] = sc[tid];
  }
#pragma unroll
  for (int off = 16; off > 0; off >>= 1)
    psum += __shfl_xor(psum, off, 32);
  if (lane == 0) red[wave] = psum;
  __syncthreads();
  if (tid == 0) {
    float s = 0.f;
    for (int w = 0; w < 16; ++w) s += red[w];
    Mc[b * NCH + ch] = m_local;
    Sc[b * NCH + ch] = s;
  }
  __syncthreads();

  // ---- Phase B: partial weighted values ---------------------------------
  float acc0 = 0.f, acc1 = 0.f;
  const int h = tid;   // dir0: d=h ; dir1: d=512+h

#if HAVE_ASYNC_LDS
  // Async-DMA a tile of TT timesteps (both directions) into LDS.
  // Tile = TT*2 rows of 512 floats = 2048 b128 chunks; 512 threads x 4 each.
  auto issue_tile = [&](int tile, int buf) {
    const int t0 = ch * TC + tile * TT;
#pragma unroll
    for (int r = 0; r < 4; ++r) {
      const int chunk = r * 512 + tid;      // 0..2047
      const int row   = chunk >> 7;         // 0..15 = t8*2 + dir
      const int pos   = (chunk & 127) * 4;  // float offset within row
      const int t8    = row >> 1;
      const int dir   = row & 1;
      const float* g = enc + (((size_t)(dir ? T_ : 0) + t0 + t8) * B_ + b) * H_ + pos;
      float*       l = &ktile[buf][(row << 9) + pos];
      __builtin_amdgcn_global_load_async_to_lds_b128(
          (gv4i*)(void*)g, (lv4i*)(void*)l, /*offset=*/0, /*cpol=*/0);
    }
  };

  int cur = 0;
  issue_tile(0, cur);
  async_wait0();
  __syncthreads();
  for (int tile = 0; tile < TC / TT; ++tile) {
    const int nxt = cur ^ 1;
    if (tile + 1 < TC / TT) issue_tile(tile + 1, nxt);  // overlap DMA w/ math
    const float* __restrict__ kt = &ktile[cur][0];
#pragma unroll
    for (int t8 = 0; t8 < TT; ++t8) {
      const float w = wts[tile * TT + t8];
      acc0 += w * kt[(t8 * 2 + 0) * H_ + h];
      acc1 += w * kt[(t8 * 2 + 1) * H_ + h];
    }
    async_wait0();     // own ASYNCcnt drained
    __syncthreads();   // all 16 waves' tile writes visible; cur free to reuse
    cur = nxt;
  }
#else
  const float* __restrict__ base0 = enc + ((size_t)(ch * TC) * B_ + b) * H_ + h;
  const float* __restrict__ base1 = enc + (((size_t)T_ + ch * TC) * B_ + b) * H_ + h;
  for (int tt = 0; tt < TC; ++tt) {
    const float w = wts[tt];
    if (tt + 4 < TC) {
      __builtin_prefetch(base0 + (size_t)(tt + 4) * B_ * H_, 0, 1);
      __builtin_prefetch(base1 + (size_t)(tt + 4) * B_ * H_, 0, 1);
    }
    acc0 += w * base0[(size_t)tt * B_ * H_];
    acc1 += w * base1[(size_t)tt * B_ * H_];
  }
#endif

  float* __restrict__ Pb = P + ((size_t)b * NCH + ch) * DH_;
  Pb[h]      = acc0;
  Pb[H_ + h] = acc1;
}

// ---------------------------------------------------------------------------
// Kernel 3: combine chunk partials -> final values + normalized scores.
// ---------------------------------------------------------------------------
__global__ void attn_combine(const float* __restrict__ Sraw,
                             const float* __restrict__ Mc,
                             const float* __restrict__ Sc,
                             const float* __restrict__ P,
                             float* __restrict__ out_val,
                             float* __restrict__ out_sc) {
  const int b   = blockIdx.x;
  const int tid = threadIdx.x;   // 0..1023
  __shared__ float w_c[NCH];
  __shared__ float m_s, inv_s;

  if (tid == 0) {
    float m = Mc[b * NCH];
    for (int c = 1; c < NCH; ++c) m = fmaxf(m, Mc[b * NCH + c]);
    float denom = 0.f;
    for (int c = 0; c < NCH; ++c) {
      const float e = __expf(Mc[b * NCH + c] - m);
      w_c[c] = e;
      denom += Sc[b * NCH + c] * e;
    }
    m_s   = m;
    inv_s = 1.f / denom;
  }
  __syncthreads();
  const float inv = inv_s;
  const float m   = m_s;

  // attn_values[b, tid]
  float acc = 0.f;
#pragma unroll
  for (int c = 0; c < NCH; ++c)
    acc += P[((size_t)b * NCH + c) * DH_ + tid] * w_c[c];
  out_val[(size_t)b * DH_ + tid] = acc * inv;

  // attn_scores[b, t, 0]
  for (int t = tid; t < T_; t += 1024)
    out_sc[(size_t)b * T_ + t] = __expf(Sraw[(size_t)b * T_ + t] - m) * inv;
}

// ---------------------------------------------------------------------------
extern "C" void kernel_launch(void* const* d_in, const int* in_sizes, int n_in,
                              void* d_out, int out_size, void* d_ws, size_t ws_size,
                              hipStream_t stream) {
  const float* inp = (const float*)d_in[0];  // (32, 1024)
  const float* enc = (const float*)d_in[1];  // (2, 2048, 32, 512)
  // d_in[2] decoder_state: unused by forward
  const float* Wq  = (const float*)d_in[3];  // (1024, 1024)
  const float* bq  = (const float*)d_in[4];  // (1024,)

  float* ws = (float*)d_ws;
  float* Q  = ws + Q_OFF;
  float* S  = ws + S_OFF;
  float* Mc = ws + M_OFF;
  float* Sc = ws + SU_OFF;
  float* P  = ws + P_OFF;

  float* out_val = (float*)d_out;            // (32, 1024)
  float* out_sc  = (float*)d_out + B_ * DH_; // (32, 2048, 1)

  qproj_wmma<<<16, 256, 0, stream>>>(inp, Wq, bq, Q);   // 128 tiles / 8 waves per block
  dim3 g2(NCH, B_);
  attn_partial<<<g2, 512, 0, stream>>>(enc, Q, S, Mc, Sc, P);
  attn_combine<<<B_, 1024, 0, stream>>>(S, Mc, Sc, P, out_val, out_sc);
}